// CGNN_20486994002407
// MI455X (gfx1250) — compile-verified
//
#include <hip/hip_runtime.h>

typedef float v2f __attribute__((ext_vector_type(2)));
typedef float v8f __attribute__((ext_vector_type(8)));

// D = A(16x4 f32) * B(4x16 f32) + C(16x16 f32)
__device__ __forceinline__ v8f wmma4(v2f a, v2f b, v8f c) {
    return __builtin_amdgcn_wmma_f32_16x16x4_f32(false, a, false, b, (short)0, c, false, false);
}

// Half-wave value select (m ? hi : lo) in bitwise form; lowers to one
// v_cndmask/v_bfi and cannot be canonicalized into a dynamic-index extract.
__device__ __forceinline__ float hsel(int m, float lo, float hi) {
    return __int_as_float((__float_as_int(hi) & m) | (__float_as_int(lo) & ~m));
}

__device__ __forceinline__ float relu(float x) {
    return __builtin_amdgcn_fmed3f(x, 0.0f, __builtin_inff());
}

// A[row, k] element; row duplication (rows 8+j mirror rows j) via fe = lane & 7.
// w is (n_in, n_out) row-major; bias folded at k == n_in.
__device__ __forceinline__ float aElem(const float* w, const float* b,
                                       int n_out, int n_in, int fe, int k) {
    if (fe >= n_out) return 0.0f;
    if (k < n_in)   return w[k * n_out + fe];
    if (k == n_in)  return b[fe];
    return 0.0f;
}

// One 3->3->6->3 MLP over all 12 windows. Input indices are (2w+S, 2w+S+1, 2w+S+2) % 24;
// S=23 reproduces IDX1 ((2w-1)%24, 2w, 2w+1), S=0 reproduces IDX2 (2w, 2w+1, (2w+2)%24).
// Kept as a separate pass so only this MLP's 4 A-operands are live at once.
template<int S>
__device__ __forceinline__ void mlp_3in_pass(
    const float* __restrict__ w1, const float* __restrict__ b1,
    const float* __restrict__ w2, const float* __restrict__ b2,
    const float* __restrict__ w3, const float* __restrict__ b3,
    const float (&u)[24], int fe, int m,
    int kLo0, int kLo1, int kHi0, int kHi1,
    bool doStore, float* __restrict__ obase)
{
    v2f A1  = { aElem(w1, b1, 3, 3, fe, kLo0), aElem(w1, b1, 3, 3, fe, kLo1) };
    v2f A2  = { aElem(w2, b2, 6, 3, fe, kLo0), aElem(w2, b2, 6, 3, fe, kLo1) };
    v2f A3a = { aElem(w3, b3, 3, 6, fe, kLo0), aElem(w3, b3, 3, 6, fe, kLo1) };
    v2f A3b = { aElem(w3, b3, 3, 6, fe, kHi0), aElem(w3, b3, 3, 6, fe, kHi1) };
    v8f zero8 = {0,0,0,0,0,0,0,0};

    #pragma unroll
    for (int w = 0; w < 12; ++w) {
        const int p0 = (2*w + S)     % 24;
        const int p1 = (2*w + S + 1) % 24;
        const int p2 = (2*w + S + 2) % 24;

        v2f bb1 = { hsel(m, u[p0], u[p2]), hsel(m, u[p1], 1.0f) };    // K3 = ones (bias)
        v8f d   = wmma4(A1, bb1, zero8);
        float h0 = relu(d[0]), h1 = relu(d[1]), h2 = relu(d[2]);
        v2f bb2 = { hsel(m, h0, h2), hsel(m, h1, 1.0f) };             // dup rows read in-lane
        v8f d2  = wmma4(A2, bb2, zero8);
        float g0 = relu(d2[0]), g1 = relu(d2[1]), g2 = relu(d2[2]),
              g3 = relu(d2[3]), g4 = relu(d2[4]), g5 = relu(d2[5]);
        v2f b3a = { hsel(m, g0, g2),   hsel(m, g1, g3) };
        v2f b3b = { hsel(m, g4, 1.0f), hsel(m, g5, 0.0f) };           // K6 = ones (bias)
        v8f d3  = wmma4(A3b, b3b, zero8);
        d3 = wmma4(A3a, b3a, d3);
        if (doStore) {
            float* o = obase + w * 3;
            o[0] = d3[0]; o[1] = d3[1]; o[2] = d3[2];
        }
    }
}

__global__ __launch_bounds__(256)
void CGNN_20486994002407_kernel(
    const float* __restrict__ u1,
    const float* __restrict__ w11, const float* __restrict__ b11,
    const float* __restrict__ w12, const float* __restrict__ b12,
    const float* __restrict__ w13, const float* __restrict__ b13,
    const float* __restrict__ w21, const float* __restrict__ b21,
    const float* __restrict__ w22, const float* __restrict__ b22,
    const float* __restrict__ w23, const float* __restrict__ b23,
    const float* __restrict__ w31, const float* __restrict__ b31,
    const float* __restrict__ w32, const float* __restrict__ b32,
    const float* __restrict__ w33, const float* __restrict__ b33,
    float* __restrict__ out1, float* __restrict__ out2, float* __restrict__ out3,
    int n)
{
    const int  lane = threadIdx.x & 31;
    const int  wave = threadIdx.x >> 5;
    const int  f    = lane & 15;           // item (N dim)
    const int  fe   = lane & 7;            // effective weight row (rows 8+j duplicate j)
    const bool hiHalf = lane >= 16;
    const int  m    = hiHalf ? -1 : 0;     // half-wave select mask

    const long rbase = ((long)blockIdx.x * (blockDim.x >> 5) + wave) * 16;
    if (rbase >= n) return;                // wave-uniform exit (EXEC all-1 for WMMA)

    const long r  = rbase + f;
    const long rc = (r < n) ? r : (long)(n - 1);   // clamp reads; stores are guarded

    // ---- load this lane's row of u1 (both half-waves hold the same row) ----
    float u[24];
    {
        const float4* up = reinterpret_cast<const float4*>(u1 + rc * 24);
        #pragma unroll
        for (int i = 0; i < 6; ++i) {
            float4 t = up[i];
            u[4*i+0] = t.x; u[4*i+1] = t.y; u[4*i+2] = t.z; u[4*i+3] = t.w;
        }
    }

    // A-operand K slots: VGPR0 holds K=0 (lanes 0-15) / K=2 (lanes 16-31); VGPR1: K=1 / K=3
    const int kLo0 = hiHalf ? 2 : 0, kLo1 = hiHalf ? 3 : 1;
    const int kHi0 = hiHalf ? 6 : 4, kHi1 = hiHalf ? 7 : 5;

    const bool doStore = (!hiHalf) && (r < n);

    // ================= pass 1: MLP1 (IDX1) =================
    mlp_3in_pass<23>(w11, b11, w12, b12, w13, b13, u, fe, m,
                     kLo0, kLo1, kHi0, kHi1, doStore, out1 + r * 36);

    // ================= pass 2: MLP2 (IDX2) =================
    mlp_3in_pass<0>(w21, b21, w22, b22, w23, b23, u, fe, m,
                    kLo0, kLo1, kHi0, kHi1, doStore, out2 + r * 36);

    // ================= pass 3: MLP3 4 -> 4 -> 6 -> 2 (IDX3) =================
    {
        v2f A1  = { aElem(w31, b31, 4, 4, fe, kLo0), aElem(w31, b31, 4, 4, fe, kLo1) };
        v2f A2  = { aElem(w32, b32, 6, 4, fe, kLo0), aElem(w32, b32, 6, 4, fe, kLo1) };
        v2f A3a = { aElem(w33, b33, 2, 6, fe, kLo0), aElem(w33, b33, 2, 6, fe, kLo1) };
        v2f A3b = { aElem(w33, b33, 2, 6, fe, kHi0), aElem(w33, b33, 2, 6, fe, kHi1) };

        // bias accumulators (K fully occupied by features); lane-uniform so the
        // duplicated rows M=v+8 get the same bias as rows M=v.
        v8f zero8 = {0,0,0,0,0,0,0,0};
        v8f c1 = zero8, c2 = zero8;
        c1[0] = b31[0]; c1[1] = b31[1]; c1[2] = b31[2]; c1[3] = b31[3];
        c2[0] = b32[0]; c2[1] = b32[1]; c2[2] = b32[2];
        c2[3] = b32[3]; c2[4] = b32[4]; c2[5] = b32[5];

        float* obase = out3 + r * 24;
        #pragma unroll
        for (int w = 0; w < 12; ++w) {
            const int p0 = (2*w)     % 24;     // IDX3: 2w, 2w+1, (2w+2)%24, (2w+3)%24
            const int p1 = (2*w + 1) % 24;
            const int p2 = (2*w + 2) % 24;
            const int p3 = (2*w + 3) % 24;

            v2f bb1 = { hsel(m, u[p0], u[p2]), hsel(m, u[p1], u[p3]) };
            v8f e   = wmma4(A1, bb1, c1);
            float t0 = relu(e[0]), t1 = relu(e[1]), t2 = relu(e[2]), t3 = relu(e[3]);
            v2f bb2 = { hsel(m, t0, t2), hsel(m, t1, t3) };
            v8f e2  = wmma4(A2, bb2, c2);
            float s0 = relu(e2[0]), s1 = relu(e2[1]), s2 = relu(e2[2]),
                  s3 = relu(e2[3]), s4 = relu(e2[4]), s5 = relu(e2[5]);
            v2f b3a = { hsel(m, s0, s2),   hsel(m, s1, s3) };
            v2f b3b = { hsel(m, s4, 1.0f), hsel(m, s5, 0.0f) };       // K6 = ones (bias)
            v8f e3  = wmma4(A3b, b3b, zero8);
            e3 = wmma4(A3a, b3a, e3);
            if (doStore) {
                float* o = obase + w * 2;
                o[0] = e3[0]; o[1] = e3[1];
            }
        }
    }
}

extern "C" void kernel_launch(void* const* d_in, const int* in_sizes, int n_in,
                              void* d_out, int out_size, void* d_ws, size_t ws_size,
                              hipStream_t stream) {
    (void)n_in; (void)out_size; (void)d_ws; (void)ws_size;
    const float* u1  = (const float*)d_in[0];
    const float* w11 = (const float*)d_in[1];  const float* b11 = (const float*)d_in[2];
    const float* w12 = (const float*)d_in[3];  const float* b12 = (const float*)d_in[4];
    const float* w13 = (const float*)d_in[5];  const float* b13 = (const float*)d_in[6];
    const float* w21 = (const float*)d_in[7];  const float* b21 = (const float*)d_in[8];
    const float* w22 = (const float*)d_in[9];  const float* b22 = (const float*)d_in[10];
    const float* w23 = (const float*)d_in[11]; const float* b23 = (const float*)d_in[12];
    const float* w31 = (const float*)d_in[13]; const float* b31 = (const float*)d_in[14];
    const float* w32 = (const float*)d_in[15]; const float* b32 = (const float*)d_in[16];
    const float* w33 = (const float*)d_in[17]; const float* b33 = (const float*)d_in[18];

    const int n = in_sizes[0] / 24;
    float* out  = (float*)d_out;
    float* out1 = out;
    float* out2 = out + (size_t)n * 36;
    float* out3 = out + (size_t)n * 72;

    const int rowsPerBlock = 128;  // 256 threads = 8 waves * 16-row tiles
    const int blocks = (n + rowsPerBlock - 1) / rowsPerBlock;
    hipLaunchKernelGGL(CGNN_20486994002407_kernel, dim3(blocks), dim3(256), 0, stream,
                       u1, w11, b11, w12, b12, w13, b13,
                           w21, b21, w22, b22, w23, b23,
                           w31, b31, w32, b32, w33, b33,
                       out1, out2, out3, n);
}